// QubitEncoder_86071144612497
// MI455X (gfx1250) — compile-verified
//
#include <hip/hip_runtime.h>

// CDNA5 / gfx1250: wave32, WMMA f32 16x16x4 used as a K=2 complex outer product.
typedef __attribute__((ext_vector_type(2))) float v2f;
typedef __attribute__((ext_vector_type(8))) float v8f;

#define NQ    20      // qubits
#define NH    10      // qubits per half
#define DHALF 1024    // 2^10
#define BATCH 8

// ---------------------------------------------------------------------------
// Kernel 1: build per-batch half-amplitude tables in d_ws.
//   H[b][h] = prod_{i=0..9}  f_i(bit_{9-i}(h))   (qubits 0..9  -> idx bits 19..10)
//   L[b][l] = prod_{i=10..19} f_i(bit_{19-i}(l)) (qubits 10..19 -> idx bits 9..0)
//   f_i(0) = cos(x_i/2),  f_i(1) = -i*sin(x_i/2)
// ws layout (floats): H = ws[0 .. 8*1024*2), L = ws[8*1024*2 .. 16*1024*2)
// ---------------------------------------------------------------------------
__global__ __launch_bounds__(DHALF) void qenc_build_tables(
    const float* __restrict__ x, float* __restrict__ ws)
{
    __shared__ float cs[NQ];
    __shared__ float sn[NQ];
    const int b = blockIdx.x;      // batch
    const int t = threadIdx.x;     // table entry 0..1023

    if (t < NQ) {
        float a = 0.5f * x[b * NQ + t];
        cs[t] = cosf(a);
        sn[t] = sinf(a);
    }
    __syncthreads();

    // hi half: qubit i (0..9) owns bit (9-i) of h
    float re = 1.0f, im = 0.0f;
    #pragma unroll
    for (int i = 0; i < NH; ++i) {
        const int bit = (t >> (9 - i)) & 1;
        const float c = cs[i], s = sn[i];
        // bit==0: z *= c ; bit==1: z *= (0,-s) -> (im*s, -re*s)
        const float nre = bit ? (im * s)  : (re * c);
        const float nim = bit ? (-re * s) : (im * c);
        re = nre; im = nim;
    }
    ws[(b * DHALF + t) * 2 + 0] = re;
    ws[(b * DHALF + t) * 2 + 1] = im;

    // lo half: qubit i (10..19) owns bit (19-i) of l
    re = 1.0f; im = 0.0f;
    #pragma unroll
    for (int i = NH; i < NQ; ++i) {
        const int bit = (t >> (19 - i)) & 1;
        const float c = cs[i], s = sn[i];
        const float nre = bit ? (im * s)  : (re * c);
        const float nim = bit ? (-re * s) : (im * c);
        re = nre; im = nim;
    }
    float* Lb = ws + BATCH * DHALF * 2;
    Lb[(b * DHALF + t) * 2 + 0] = re;
    Lb[(b * DHALF + t) * 2 + 1] = im;
}

// ---------------------------------------------------------------------------
// Kernel 2: out[(h*1024+l)*8 + b] = H[b][h] * L[b][l]  (complex outer product)
// via V_WMMA_F32_16X16X4_F32, K=2 effective:
//   re: A=[hr,hi], B=[lr,-li] ;  im: A=[hr,hi], B=[li,lr]
// Block = 256 threads = 8 waves; wave w <-> batch w, all on the same 16x16 tile
// so the 8 batches interleave into contiguous 1 KB output spans per tile row.
// Grid = (64,64) tiles over (l,h).
// ---------------------------------------------------------------------------
__global__ __launch_bounds__(256) void qenc_outer_wmma(
    const float* __restrict__ ws, float* __restrict__ out)
{
    const int lane = threadIdx.x & 31;
    const int b    = threadIdx.x >> 5;       // wave index == batch
    const int l0   = blockIdx.x * 16;
    const int h0   = blockIdx.y * 16;

    const bool lo16 = lane < 16;             // lanes 16..31 carry K=2,3 -> zero
    const int  m    = lane & 15;

    // Unconditional b64 loads (always in-bounds; upper lanes re-read m=0..15),
    // then mask with selects -> no exec-mask branches around the loads.
    const float2 Hv = *reinterpret_cast<const float2*>(
        ws + (size_t)(b * DHALF + h0 + m) * 2);
    const float2 Lv = *reinterpret_cast<const float2*>(
        ws + (size_t)(BATCH * DHALF + b * DHALF + l0 + m) * 2);

    const float hr = lo16 ? Hv.x : 0.0f;
    const float hi = lo16 ? Hv.y : 0.0f;
    const float lr = lo16 ? Lv.x : 0.0f;
    const float li = lo16 ? Lv.y : 0.0f;

    // A: 16x4 f32 -> lanes0-15: VGPR0=K0, VGPR1=K1 ; lanes16-31 (K2,K3) = 0
    v2f A;   A.x   = hr;  A.y   = hi;
    // B: 4x16 f32 -> lanes0-15 hold K0/K1 rows in VGPR0/1 ; upper half = 0
    v2f Bre; Bre.x = lr;  Bre.y = -li;
    v2f Bim; Bim.x = li;  Bim.y = lr;

    v8f cre = {};
    v8f cim = {};
    // (neg_a, A, neg_b, B, c_mod, C, reuse_a, reuse_b)
    cre = __builtin_amdgcn_wmma_f32_16x16x4_f32(false, A, false, Bre,
                                                (short)0, cre, false, false);
    cim = __builtin_amdgcn_wmma_f32_16x16x4_f32(false, A, false, Bim,
                                                (short)0, cim, false, false);

    // D layout: VGPR r -> row r (lanes0-15) / row 8+r (lanes16-31), N = lane&15
    const int n    = lane & 15;
    const int mofs = (lane >> 4) * 8;
    #pragma unroll
    for (int r = 0; r < 8; ++r) {
        const int h = h0 + r + mofs;
        const int l = l0 + n;
        const size_t idx = (size_t)h * (DHALF) + (size_t)l;   // state index
        float2 v; v.x = cre[r]; v.y = cim[r];
        *reinterpret_cast<float2*>(out + (idx * BATCH + b) * 2) = v;
    }
}

extern "C" void kernel_launch(void* const* d_in, const int* in_sizes, int n_in,
                              void* d_out, int out_size, void* d_ws, size_t ws_size,
                              hipStream_t stream)
{
    const float* x  = (const float*)d_in[0];   // (BATCH, NQ) float32 angles
    float*       out = (float*)d_out;          // (2^20, 8) complex64 -> float pairs
    float*       ws  = (float*)d_ws;           // 16*1024*2 floats = 128 KB used

    qenc_build_tables<<<dim3(BATCH), dim3(DHALF), 0, stream>>>(x, ws);
    qenc_outer_wmma<<<dim3(64, 64), dim3(256), 0, stream>>>(ws, out);
}